// PersistentEntropyLoss_81003083202857
// MI455X (gfx1250) — compile-verified
//
#include <hip/hip_runtime.h>
#include <hip/hip_bf16.h>
#include <math.h>

// ---------------------------------------------------------------------------
// PersistentEntropyLoss on MI455X (gfx1250)
//   1) rownorm_kernel : sq[i] = ||x_i||^2                    (8 MB read)
//   2) dist_gemm      : D = sqrt(max(sq_i+sq_j-2*X.X^T,0)+eps) via
//                       V_WMMA_F32_16X16X4_F32 (exact f32 matrix pipe)
//   3) prim_entropy   : sequential Prim MST in one 1024-thread workgroup,
//                       mindist in LDS, closed-form persistent entropy.
// Workspace: D = 8192*8192 f32 (256 MB) + sq = 8192 f32.
// ---------------------------------------------------------------------------

typedef float v2f __attribute__((ext_vector_type(2)));
typedef float v8f __attribute__((ext_vector_type(8)));

#define NPTS 8192
#define DFEAT 256

// ---------------- row squared norms: one wave per row ----------------------
__global__ __launch_bounds__(256) void rownorm_kernel(const float* __restrict__ X,
                                                      float* __restrict__ sq) {
    int wave = threadIdx.x >> 5;
    int lane = threadIdx.x & 31;
    int row  = blockIdx.x * 8 + wave;
    const float4* xr = reinterpret_cast<const float4*>(X + (size_t)row * DFEAT);
    float4 a = xr[lane * 2 + 0];
    float4 b = xr[lane * 2 + 1];
    float s = a.x * a.x + a.y * a.y + a.z * a.z + a.w * a.w
            + b.x * b.x + b.y * b.y + b.z * b.z + b.w * b.w;
#pragma unroll
    for (int off = 16; off > 0; off >>= 1) s += __shfl_down(s, off);
    if (lane == 0) sq[row] = s;
}

// ---------------- distance GEMM via f32 WMMA -------------------------------
// Block: 256 threads = 8 waves (2x4), 128x128 output tile.
// Wave: 64x32 region = 4x2 tiles of 16x16, K-loop over 256 in steps of 4.
__global__ __launch_bounds__(256) void dist_gemm_kernel(const float* __restrict__ X,
                                                        const float* __restrict__ sq,
                                                        float* __restrict__ D) {
    const int lane  = threadIdx.x & 31;
    const int wave  = threadIdx.x >> 5;
    const int waveM = wave >> 2;      // 0..1
    const int waveN = wave & 3;       // 0..3
    const int l15   = lane & 15;
    const int khalf = lane >> 4;      // 0: K=0,1  1: K=2,3  (ISA 16x4 f32 layout)

    const int rowBase = blockIdx.y * 128 + waveM * 64;
    const int colBase = blockIdx.x * 128 + waveN * 32;

    v8f acc[4][2] = {};

    const float* aPtr[4];
    const float* bPtr[2];
#pragma unroll
    for (int mt = 0; mt < 4; ++mt)
        aPtr[mt] = X + (size_t)(rowBase + mt * 16 + l15) * DFEAT + khalf * 2;
#pragma unroll
    for (int nt = 0; nt < 2; ++nt)
        bPtr[nt] = X + (size_t)(colBase + nt * 16 + l15) * DFEAT + khalf * 2;

#pragma unroll 4
    for (int kb = 0; kb < DFEAT; kb += 4) {
        v2f a[4], b[2];
#pragma unroll
        for (int mt = 0; mt < 4; ++mt)
            a[mt] = *reinterpret_cast<const v2f*>(aPtr[mt] + kb);
#pragma unroll
        for (int nt = 0; nt < 2; ++nt)
            b[nt] = *reinterpret_cast<const v2f*>(bPtr[nt] + kb);
#pragma unroll
        for (int mt = 0; mt < 4; ++mt)
#pragma unroll
            for (int nt = 0; nt < 2; ++nt)
                acc[mt][nt] = __builtin_amdgcn_wmma_f32_16x16x4_f32(
                    false, a[mt], false, b[nt], (short)0, acc[mt][nt], false, false);
    }

    // Epilogue: g -> sqrt(max(sqi+sqj-2g,0)+1e-12), store D.
    // C layout: VGPR r, lanes0-15: M=r, N=lane; lanes16-31: M=8+r, N=lane-16.
#pragma unroll
    for (int mt = 0; mt < 4; ++mt) {
#pragma unroll
        for (int nt = 0; nt < 2; ++nt) {
            const int col = colBase + nt * 16 + l15;
            const float sqc = sq[col];
#pragma unroll
            for (int r = 0; r < 8; ++r) {
                const int row = rowBase + mt * 16 + khalf * 8 + r;
                float g  = acc[mt][nt][r];
                float d2 = sq[row] + sqc - 2.0f * g;
                d2 = fmaxf(d2, 0.0f);
                D[(size_t)row * NPTS + col] = sqrtf(d2 + 1e-12f);
            }
        }
    }
}

// ---------------- Prim's MST + persistent entropy (one workgroup) ----------
__global__ __launch_bounds__(1024) void prim_entropy_kernel(const float* __restrict__ D,
                                                            float* __restrict__ out) {
    const int T = 1024;
    __shared__ float mind[NPTS];              // 32 KB
    __shared__ unsigned int intree[NPTS / 32]; // 1 KB bitmask
    __shared__ float redV[32];
    __shared__ int   redI[32];
    __shared__ int   curJ;

    const int tid = threadIdx.x;

    for (int k = tid; k < NPTS; k += T) mind[k] = D[k];      // row 0
    for (int k = tid; k < NPTS / 32; k += T) intree[k] = 0u;
    if (tid == 0) intree[0] = 1u;
    __syncthreads();

    float s0 = 0.0f, s1 = 0.0f;   // only meaningful on tid 0

    for (int it = 0; it < NPTS - 1; ++it) {
        // fused scan + local argmin over owned elements (k = tid + i*1024)
        float bv = 3.0e38f;
        int   bi = NPTS;
        for (int k = tid; k < NPTS; k += T) {
            if (!((intree[k >> 5] >> (k & 31)) & 1u)) {
                float v = mind[k];
                if (v < bv) { bv = v; bi = k; }   // strict < => first index on ties
            }
        }
        // wave argmin (lower index wins ties, matching jnp.argmin)
#pragma unroll
        for (int off = 16; off > 0; off >>= 1) {
            float ov = __shfl_down(bv, off);
            int   oi = __shfl_down(bi, off);
            if (ov < bv || (ov == bv && oi < bi)) { bv = ov; bi = oi; }
        }
        if ((tid & 31) == 0) { redV[tid >> 5] = bv; redI[tid >> 5] = bi; }
        __syncthreads();
        if (tid < 32) {
            bv = redV[tid];
            bi = redI[tid];
#pragma unroll
            for (int off = 16; off > 0; off >>= 1) {
                float ov = __shfl_down(bv, off);
                int   oi = __shfl_down(bi, off);
                if (ov < bv || (ov == bv && oi < bi)) { bv = ov; bi = oi; }
            }
            if (tid == 0) {
                curJ = bi;
                intree[bi >> 5] |= (1u << (bi & 31));
                s0 += bv;
                s1 += bv * logf(bv);   // bv >= sqrt(1e-12) > 0
            }
        }
        __syncthreads();
        // relax with row j (coalesced 32 KB read, mostly L2-resident)
        const int j = curJ;
        const float* __restrict__ rowj = D + (size_t)j * NPTS;
        for (int k = tid; k < NPTS; k += T)
            mind[k] = fminf(mind[k], rowj[k]);
        // no extra barrier needed: each thread re-reads only its own mind[k],
        // and intree/curJ updates are fenced by the two barriers above.
    }

    if (tid == 0) {
        // E = -sum p log(p+eps), p = w/(S0+eps); eps negligible inside log in f32
        // loss = -E = (S1 - S0*log(Tt))/Tt ; out = WEIGHT * loss
        float Tt = s0 + 1e-12f;
        out[0] = (s1 - s0 * logf(Tt)) / Tt;
    }
}

// ---------------------------------------------------------------------------
extern "C" void kernel_launch(void* const* d_in, const int* in_sizes, int n_in,
                              void* d_out, int out_size, void* d_ws, size_t ws_size,
                              hipStream_t stream) {
    (void)in_sizes; (void)n_in; (void)out_size; (void)ws_size;

    const float* X = (const float*)d_in[0];      // [8192, 256] f32
    float* out = (float*)d_out;                  // scalar f32

    float* D  = (float*)d_ws;                    // 8192*8192 f32 = 256 MB
    float* sq = D + (size_t)NPTS * NPTS;         // 8192 f32

    rownorm_kernel<<<NPTS / 8, 256, 0, stream>>>(X, sq);

    dim3 grid(NPTS / 128, NPTS / 128);           // 64 x 64 blocks
    dist_gemm_kernel<<<grid, 256, 0, stream>>>(X, sq, D);

    prim_entropy_kernel<<<1, 1024, 0, stream>>>(D, out);
}